// RopeEmbeddings_77635828842545
// MI455X (gfx1250) — compile-verified
//
#include <hip/hip_runtime.h>

// RoPE, fp32, X:(L=2048, D=4096, N=4) row-major.
// Streaming op: 256 MiB traffic -> ~11.5us floor @ 23.3 TB/s. Strategy:
//  - tile (16 l x 64 j) per block; theta_j computed once per block (double
//    exp2, correctly rounded to fp32) and shared via LDS -> exp2 amortized 16x
//  - custom branchless sincos: double-precision pi/2 reduction (input is a
//    known-exact fp32 in [1, 2.1e7]) + fp32 minimax polys (~2 ulp)
//  - b128 non-temporal loads/stores (footprint > 192MB L2, zero reuse)
//  - global_prefetch one row-group ahead

typedef __attribute__((ext_vector_type(4))) float v4f;

#define L_DIM 2048
#define D_DIM 4096
#define N_DIM 4
#define HALF  (D_DIM / 2)        // 2048
#define TL    16                 // l-rows per block
#define TJ    64                 // j-cols per block
#define TPB   256                // 8 wave32
#define ITERS ((TL * TJ) / TPB)  // 4
#define JBLOCKS (HALF / TJ)      // 32
#define LBLOCKS (L_DIM / TL)     // 128
#define PF_ELEMS (4 * D_DIM * N_DIM)  // next iteration is 4 l-rows ahead

// 2*log2(10000)/D, folded in double at compile time.
#define THETA_C (2.0 * 13.287712379549449 / (double)D_DIM)

// sin/cos of the exact fp32 value ang, ang in [1, ~2.1e7].
__device__ __forceinline__ void rope_sincos(float ang, float* sr, float* cr) {
  const double TWO_OVER_PI = 0.63661977236758134308;
  const double PIO2_HI = 1.5707963267948966;      // double(pi/2)
  const double PIO2_LO = 6.123233995736766e-17;   // pi/2 - PIO2_HI
  double d = (double)ang;
  double k = rint(d * TWO_OVER_PI);               // k in [0, 1.31e7]
  double t = fma(-k, PIO2_HI, d);
  double r = fma(-k, PIO2_LO, t);                 // |r| <= pi/4, err ~2e-16
  float rf = (float)r;
  int iq = (int)k & 3;
  float r2 = rf * rf;
  // fp32 minimax on [-pi/4, pi/4] (cephes coefficients)
  float ps = fmaf(r2, fmaf(r2, -1.9515295891e-4f, 8.3321608736e-3f),
                  -1.6666654611e-1f);
  float s = fmaf(rf * r2, ps, rf);
  float pc = fmaf(r2, fmaf(r2, 2.443315711809948e-5f, -1.388731625493765e-3f),
                  4.166664568298827e-2f);
  float c = fmaf(r2 * r2, pc, fmaf(r2, -0.5f, 1.0f));
  // quadrant: sin -> {s, c, -s, -c}[iq], cos -> {c, -s, -c, s}[iq]
  bool swap = (iq & 1) != 0;
  float ss = swap ? c : s;
  float cc = swap ? s : c;
  ss = (iq & 2) ? -ss : ss;
  cc = ((iq + 1) & 2) ? -cc : cc;
  *sr = ss;
  *cr = cc;
}

__global__ __launch_bounds__(TPB) void rope_f32_kernel(
    const float* __restrict__ X, float* __restrict__ out) {
  __shared__ float sTheta[TJ];

  const int bj = blockIdx.x % JBLOCKS;
  const int bl = blockIdx.x / JBLOCKS;
  const int j0 = bj * TJ;
  const int l0 = bl * TL;

  // Phase 1: 64 threads compute theta_j = 10000^(2j/D), correctly rounded
  // to fp32 via double exp2; amortized over the 16 l-rows of the tile.
  if (threadIdx.x < TJ) {
    const int j = j0 + threadIdx.x;
    sTheta[threadIdx.x] = (float)exp2((double)j * THETA_C);
  }
  __syncthreads();

  // Phase 2: each wave32 sweeps 32 consecutive j in one l-row per iteration
  // -> 512B contiguous b128 traffic per wave per stream.
#pragma unroll
  for (int iter = 0; iter < ITERS; ++iter) {
    const int p = iter * TPB + threadIdx.x;   // [0, 1024)
    const int row = p >> 6;                   // l offset in tile
    const int col = p & (TJ - 1);             // j offset in tile
    const int l = l0 + row;
    const int j = j0 + col;

    const size_t off1 = ((size_t)(l * D_DIM + j)) * (size_t)N_DIM;
    const size_t off2 = off1 + (size_t)HALF * (size_t)N_DIM;

    if (iter < ITERS - 1) {
      __builtin_prefetch(X + off1 + PF_ELEMS, 0, 3);
      __builtin_prefetch(X + off2 + PF_ELEMS, 0, 3);
    }

    const float theta = sTheta[col];
    const float ang = (float)(l + 1) * theta;   // reference's single fp32 mul
    float s, c;
    rope_sincos(ang, &s, &c);

    const v4f x1 = __builtin_nontemporal_load((const v4f*)(X + off1));
    const v4f x2 = __builtin_nontemporal_load((const v4f*)(X + off2));

    const v4f o1 = c * x1 - s * x2;
    const v4f o2 = s * x1 + c * x2;

    __builtin_nontemporal_store(o1, (v4f*)(out + off1));
    __builtin_nontemporal_store(o2, (v4f*)(out + off2));
  }
}

extern "C" void kernel_launch(void* const* d_in, const int* in_sizes, int n_in,
                              void* d_out, int out_size, void* d_ws, size_t ws_size,
                              hipStream_t stream) {
  (void)in_sizes; (void)n_in; (void)out_size; (void)d_ws; (void)ws_size;
  const float* X = (const float*)d_in[0];
  float* out = (float*)d_out;
  hipLaunchKernelGGL(rope_f32_kernel, dim3(LBLOCKS * JBLOCKS), dim3(TPB), 0,
                     stream, X, out);
}